// MultiHeadGravitationalAttention_32993938768207
// MI455X (gfx1250) — compile-verified
//
#include <hip/hip_runtime.h>
#include <hip/hip_bf16.h>
#include <math.h>

// MultiHeadGravitationalAttention for MI455X (gfx1250, wave32).
// All GEMM-like work goes through V_WMMA_F32_16X16X4_F32 (CDNA5's fp32 matrix op).
//
// Pipeline:
//   k_mass : masses[b,h,s] = softplus(x_head . W_mass[h])
//   k_norm : norms[s] = ||pos[s]||^2
//   k_dist : rdist[i,j] = 1 / warp(||pi||^2+||pj||^2 - 2 pi.pj)  (WMMA gram;
//            storing the RECIPROCAL kills 134M divides in the attention loop)
//   k_attn : ctx = softmax(min(|G| m_i m_j * rdist, 50)) @ V  (WMMA P.V; no-max
//            softmax is safe since forces in [0,50] -> exp in [1, 5.2e21] fp32)
//   k_out  : out = ctx @ W_out^T                                (WMMA)

typedef __attribute__((ext_vector_type(2))) float v2f;
typedef __attribute__((ext_vector_type(8))) float v8f;

#define SEQ 2048
#define DMODEL 1024
#define NHEAD 16
#define HDIM 64
#define DPOS 64
#define NBATCH 2

__device__ __forceinline__ v8f wmma_f32(v2f a, v2f b, v8f c) {
  // 8 args: (neg_a, A, neg_b, B, c_mod, C, reuse_a, reuse_b)
  return __builtin_amdgcn_wmma_f32_16x16x4_f32(false, a, false, b, (short)0, c,
                                               false, false);
}

// ---------------------------------------------------------------- masses ----
__global__ void k_mass(const float* __restrict__ x, const float* __restrict__ Wm,
                       float* __restrict__ masses) {
  int tid = blockIdx.x * blockDim.x + threadIdx.x;     // B*H*S = 65536
  int s = tid & (SEQ - 1);
  int h = (tid >> 11) & (NHEAD - 1);
  int b = tid >> 15;
  const float* xr = x + ((size_t)(b * SEQ + s)) * DMODEL + h * HDIM;
  const float* wr = Wm + h * HDIM;
  float acc = 0.f;
#pragma unroll 8
  for (int k = 0; k < HDIM; ++k) acc = fmaf(xr[k], wr[k], acc);
  masses[tid] = (acc > 20.f) ? acc : log1pf(__expf(acc));   // softplus
}

// ------------------------------------------------------------- pos norms ----
__global__ void k_norm(const float* __restrict__ pos, float* __restrict__ norms) {
  int s = blockIdx.x * blockDim.x + threadIdx.x;       // S = 2048
  const float* p = pos + (size_t)s * DPOS;
  float acc = 0.f;
#pragma unroll 8
  for (int k = 0; k < DPOS; ++k) acc = fmaf(p[k], p[k], acc);
  norms[s] = acc;
}

// ------------------------------------ warped 1/distance^2 (WMMA gram) -------
// One wave per 16x16 tile of the 2048x2048 matrix.
__global__ void k_dist(const float* __restrict__ pos, const float* __restrict__ norms,
                       float* __restrict__ rdist) {
  int wave = (blockIdx.x * blockDim.x + threadIdx.x) >> 5;   // 16384 tiles
  int lane = threadIdx.x & 31;
  int tj = wave & 127, ti = wave >> 7;
  int i0 = ti * 16, j0 = tj * 16;
  int m = lane & 15;
  int koff = (lane >> 4) << 1;                  // K sub-offset for this half-wave

  const float* pa = pos + (size_t)(i0 + m) * DPOS;
  const float* pb = pos + (size_t)(j0 + m) * DPOS;
  v8f c = {};
#pragma unroll 4
  for (int kk = 0; kk < 16; ++kk) {             // K = 64, 4 per WMMA
    int k = kk * 4 + koff;
    v2f a; a.x = pa[k]; a.y = pa[k + 1];        // A[m][k..k+1]
    v2f bb; bb.x = pb[k]; bb.y = pb[k + 1];     // B[k..k+1][n] = pos[j0+n][k..]
    c = wmma_f32(a, bb, c);                     // gram += P_i . P_j
  }
  int half8 = (lane >> 4) << 3;
  float nj = norms[j0 + m];
#pragma unroll
  for (int r = 0; r < 8; ++r) {
    int i = i0 + half8 + r;
    float d = norms[i] + nj - 2.f * c[r];
    d = fmaxf(d, 0.f);                          // kill cancellation negatives
    float dn = __fsqrt_rn(d + 1e-6f);
    d = d * (1.f + 0.15f * __cosf(dn));         // curvature warp
    d = fmaxf(d, 1e-6f);                        // event horizon clamp
    rdist[(size_t)i * SEQ + (j0 + m)] = 1.f / d;  // reciprocal, amortized B*H x
  }
}

// ----------------------------------------- attention (no-max softmax) -------
// One wave per (b, h, 16-row i-tile): stream 128 j-tiles, build the 16x16
// probability tile directly in WMMA A-fragment layout, accumulate P.V into
// four 16x16 f32 accumulators (N = hd = 64), normalize at the end.
__global__ void k_attn(const float* __restrict__ x, const float* __restrict__ masses,
                       const float* __restrict__ G, const float* __restrict__ rdist,
                       float* __restrict__ ctx) {
  int wave = (blockIdx.x * blockDim.x + threadIdx.x) >> 5;   // 4096 tasks
  int lane = threadIdx.x & 31;
  int it = wave & 127;
  int h = (wave >> 7) & 15;
  int b = wave >> 11;
  int i0 = it * 16;
  int m = lane & 15;
  int koff = (lane >> 4) << 1;

  const float* mrow = masses + ((size_t)(b * NHEAD + h) << 11);   // [S]
  const float* xb = x + (size_t)b * SEQ * DMODEL + h * HDIM;      // V base
  const float* drow = rdist + (size_t)(i0 + m) * SEQ;
  float ci = fabsf(G[h]) * mrow[i0 + m];       // row coefficient |G_h| * m_i

  v8f cacc[4] = {};                            // 16 rows x 64 cols of output
  float rsum = 0.f;
  for (int jt = 0; jt < 128; ++jt) {
    int j0 = jt * 16;
    if (jt < 127) {                            // pull next L2-resident tile in
      __builtin_prefetch(drow + j0 + 16 + koff, 0, 0);
      __builtin_prefetch(mrow + j0 + 16 + koff, 0, 0);
    }
    v2f a[4];
#pragma unroll
    for (int kk = 0; kk < 4; ++kk) {           // build P tile, A-layout
      int j = j0 + kk * 4 + koff;
      float w0 = __expf(fminf(ci * mrow[j] * drow[j], 50.f));
      float w1 = __expf(fminf(ci * mrow[j + 1] * drow[j + 1], 50.f));
      a[kk].x = w0; a[kk].y = w1;
      rsum += w0 + w1;                         // forces>=0 -> exp>=1, no underflow
    }
    // kk outer so 4 consecutive WMMAs share SRC0 (A operand reuse path)
#pragma unroll
    for (int kk = 0; kk < 4; ++kk) {
      int jr = j0 + kk * 4 + koff;
#pragma unroll
      for (int nc = 0; nc < 4; ++nc) {
        v2f bb;                                // B[k][n] = V[j0+k][nc*16+n]
        bb.x = xb[(size_t)jr * DMODEL + nc * 16 + m];
        bb.y = xb[(size_t)(jr + 1) * DMODEL + nc * 16 + m];
        cacc[nc] = wmma_f32(a[kk], bb, cacc[nc]);
      }
    }
  }
  // combine the two half-row partial sums (wave32 xor-shuffle)
  rsum += __shfl_xor(rsum, 16, 32);
  int half8 = (lane >> 4) << 3;
  float inv[8];
#pragma unroll
  for (int r = 0; r < 8; ++r)
    inv[r] = 1.f / __shfl(rsum, half8 + r, 32);   // bpermute broadcast per row
#pragma unroll
  for (int nc = 0; nc < 4; ++nc)
#pragma unroll
    for (int r = 0; r < 8; ++r)
      ctx[(size_t)(b * SEQ + i0 + half8 + r) * DMODEL + h * HDIM + nc * 16 + m] =
          cacc[nc][r] * inv[r];
}

// ------------------------------------------------- out = ctx @ W_out^T ------
// One wave per (16-row M tile, 64-col N chunk): 4096 tasks.
__global__ void k_out(const float* __restrict__ ctx, const float* __restrict__ W,
                      float* __restrict__ out) {
  int wave = (blockIdx.x * blockDim.x + threadIdx.x) >> 5;
  int lane = threadIdx.x & 31;
  int nch = wave & 15;              // 16 chunks of 64 columns
  int mt = wave >> 4;               // 256 row tiles
  int m = lane & 15;
  int koff = (lane >> 4) << 1;
  int n0 = nch * 64;

  const float* arow = ctx + (size_t)(mt * 16 + m) * DMODEL;
  v8f cacc[4] = {};
  for (int k0 = 0; k0 < DMODEL; k0 += 4) {
    v2f a; a.x = arow[k0 + koff]; a.y = arow[k0 + koff + 1];
#pragma unroll
    for (int nc = 0; nc < 4; ++nc) {
      const float* wr = W + (size_t)(n0 + nc * 16 + m) * DMODEL;
      v2f bb; bb.x = wr[k0 + koff]; bb.y = wr[k0 + koff + 1];  // B[k][n]=W[n][k]
      cacc[nc] = wmma_f32(a, bb, cacc[nc]);
    }
  }
  int half8 = (lane >> 4) << 3;
#pragma unroll
  for (int nc = 0; nc < 4; ++nc)
#pragma unroll
    for (int r = 0; r < 8; ++r)
      out[(size_t)(mt * 16 + half8 + r) * DMODEL + n0 + nc * 16 + m] =
          cacc[nc][r];
}

// ---------------------------------------------------------------------------
extern "C" void kernel_launch(void* const* d_in, const int* in_sizes, int n_in,
                              void* d_out, int out_size, void* d_ws, size_t ws_size,
                              hipStream_t stream) {
  const float* x = (const float*)d_in[0];       // [2,2048,1024]
  const float* pos = (const float*)d_in[1];     // [2048,64]
  const float* Wm = (const float*)d_in[2];      // [16,64]
  const float* G = (const float*)d_in[3];       // [16]
  const float* Wout = (const float*)d_in[4];    // [1024,1024]
  float* out = (float*)d_out;                   // [2,2048,1024]

  float* ws = (float*)d_ws;
  float* masses = ws;                                   // 65536 floats
  float* norms = ws + 65536;                            // 2048
  float* rdist = ws + 65536 + 2048;                     // 4194304 (16 MB, L2-resident)
  float* ctx = ws + 65536 + 2048 + 4194304;             // 4194304

  k_mass<<<256, 256, 0, stream>>>(x, Wm, masses);       // 65536 threads
  k_norm<<<8, 256, 0, stream>>>(pos, norms);            // 2048 threads
  k_dist<<<2048, 256, 0, stream>>>(pos, norms, rdist);  // 16384 waves
  k_attn<<<512, 256, 0, stream>>>(x, masses, G, rdist, ctx);  // 4096 waves
  k_out<<<512, 256, 0, stream>>>(ctx, Wout, out);       // 4096 waves
}